// GCNModel_78297253806422
// MI455X (gfx1250) — compile-verified
//
#include <hip/hip_runtime.h>
#include <hip/hip_bf16.h>

typedef __attribute__((ext_vector_type(2))) float v2f;
typedef __attribute__((ext_vector_type(8))) float v8f;

#define KDIM 64  // inner dimension of both GEMMs

// ---------------------------------------------------------------------------
// GEMM: out[nrows x NCOLS] = A[nrows x 64] * W[64 x NCOLS] using
// V_WMMA_F32_16X16X4_F32. One wave computes a 16-row strip (NT n-tiles).
// A-frag (16x4 f32): lanes 0-15 hold row M=lane, K={k0,k0+1};
//                    lanes 16-31 hold row M=lane-16, K={k0+2,k0+3}.
// B-frag (4x16):     lanes 0-15 col N=lane, K={k0,k0+1}; lanes 16-31 K={k0+2,k0+3}.
// C/D (16x16 f32):   VGPR i -> row (i + 8*(lane>=16)), col lane&15.
// ---------------------------------------------------------------------------
template <int NCOLS>
__global__ void gemm_wmma_f32(const float* __restrict__ A,
                              const float* __restrict__ W,
                              float* __restrict__ out, int nrows) {
  const int gtid = blockIdx.x * blockDim.x + threadIdx.x;
  const int wave = gtid >> 5;
  const int lane = threadIdx.x & 31;
  const int r0 = wave * 16;
  if (r0 >= nrows) return;  // whole-wave guard: EXEC stays all-ones for WMMA

  const int half = lane >> 4;  // 0 or 1
  const int l16 = lane & 15;
  constexpr int NT = NCOLS / 16;

  const v8f zero = {0.f, 0.f, 0.f, 0.f, 0.f, 0.f, 0.f, 0.f};
  v8f acc[NT];
#pragma unroll
  for (int t = 0; t < NT; ++t) acc[t] = zero;

  const float* arow = A + (size_t)(r0 + l16) * KDIM + half * 2;

#pragma unroll
  for (int k0 = 0; k0 < KDIM; k0 += 4) {
    v2f a;
    a.x = arow[k0 + 0];
    a.y = arow[k0 + 1];
#pragma unroll
    for (int t = 0; t < NT; ++t) {
      const float* bp = W + (size_t)(k0 + half * 2) * NCOLS + t * 16 + l16;
      v2f b;
      b.x = bp[0];
      b.y = bp[NCOLS];
      // 8 args: (neg_a, A, neg_b, B, c_mod, C, reuse_a, reuse_b)
      acc[t] = __builtin_amdgcn_wmma_f32_16x16x4_f32(
          false, a, false, b, (short)0, acc[t], false, false);
    }
  }

#pragma unroll
  for (int t = 0; t < NT; ++t) {
    float* orow = out + (size_t)r0 * NCOLS + t * 16 + l16;
#pragma unroll
    for (int i = 0; i < 8; ++i)
      orow[(size_t)(i + half * 8) * NCOLS] = acc[t][i];
  }
}

// ---------------------------------------------------------------------------
// Degree / normalization
// ---------------------------------------------------------------------------
__global__ void deg_init(float* __restrict__ deg, int n) {
  int i = blockIdx.x * blockDim.x + threadIdx.x;
  if (i < n) deg[i] = 1.0f;  // self-loop contributes 1
}

__global__ void deg_scatter(const int* __restrict__ dst, float* __restrict__ deg,
                            int E) {
  int e = blockIdx.x * blockDim.x + threadIdx.x;
  if (e < E) atomicAdd(&deg[dst[e]], 1.0f);
}

__global__ void deg_to_dinv(float* __restrict__ deg, int n) {
  int i = blockIdx.x * blockDim.x + threadIdx.x;
  if (i < n) deg[i] = rsqrtf(deg[i]);  // deg >= 1 always (self-loops)
}

// ---------------------------------------------------------------------------
// Self-loop contribution initializes the accumulator: out = dinv[i]^2 * feat
// ---------------------------------------------------------------------------
template <int F>
__global__ void self_loop_init(const float* __restrict__ feat,
                               const float* __restrict__ dinv,
                               float* __restrict__ out, int n) {
  int gid = blockIdx.x * blockDim.x + threadIdx.x;
  if (gid >= n * F) return;
  int i = gid / F;
  float w = dinv[i];
  out[gid] = w * w * feat[gid];
}

// ---------------------------------------------------------------------------
// Edge scatter: out[dst] += dinv[src]*dinv[dst] * feat[src]
// F/4 threads per edge, float4 gather, 4x GLOBAL_ATOMIC_ADD_F32 each.
// ---------------------------------------------------------------------------
template <int F>
__global__ void edge_scatter(const int* __restrict__ src,
                             const int* __restrict__ dst,
                             const float* __restrict__ dinv,
                             const float* __restrict__ feat,
                             float* __restrict__ out, int E) {
  constexpr int TPE = F / 4;
  int gid = blockIdx.x * blockDim.x + threadIdx.x;
  int e = gid / TPE;
  if (e >= E) return;
  int f4 = (gid % TPE) * 4;
  int s = src[e];
  int d = dst[e];
  float w = dinv[s] * dinv[d];
  const float4 v = *(const float4*)(feat + (size_t)s * F + f4);
  float* o = out + (size_t)d * F + f4;
  atomicAdd(o + 0, w * v.x);
  atomicAdd(o + 1, w * v.y);
  atomicAdd(o + 2, w * v.z);
  atomicAdd(o + 3, w * v.w);
}

// ---------------------------------------------------------------------------
// Bias (+ optional ReLU)
// ---------------------------------------------------------------------------
template <int F, bool RELU>
__global__ void bias_act(float* __restrict__ h, const float* __restrict__ b,
                         int n) {
  int gid = blockIdx.x * blockDim.x + threadIdx.x;
  if (gid >= n * F) return;
  float v = h[gid] + b[gid & (F - 1)];
  if (RELU) v = fmaxf(v, 0.0f);
  h[gid] = v;
}

// ---------------------------------------------------------------------------
// Launch
// ---------------------------------------------------------------------------
extern "C" void kernel_launch(void* const* d_in, const int* in_sizes, int n_in,
                              void* d_out, int out_size, void* d_ws,
                              size_t ws_size, hipStream_t stream) {
  const float* x = (const float*)d_in[0];
  const int* ei = (const int*)d_in[1];
  const float* W1 = (const float*)d_in[2];
  const float* b1 = (const float*)d_in[3];
  const float* W2 = (const float*)d_in[4];
  const float* b2 = (const float*)d_in[5];
  float* out = (float*)d_out;

  const int N = in_sizes[0] / 64;  // 100000
  const int E = in_sizes[1] / 2;   // 1600000
  const int* src = ei;
  const int* dst = ei + E;

  // workspace layout: xw[N*64] | h[N*64] | dinv[N]; hw (N*32) reuses xw
  float* xw = (float*)d_ws;
  float* h = xw + (size_t)N * 64;
  float* dinv = h + (size_t)N * 64;
  float* hw = xw;

  const int B = 256;
  auto blocks = [](long long work, int b) { return (int)((work + b - 1) / b); };

  // --- normalization coefficients ---
  deg_init<<<blocks(N, B), B, 0, stream>>>(dinv, N);
  deg_scatter<<<blocks(E, B), B, 0, stream>>>(dst, dinv, E);
  deg_to_dinv<<<blocks(N, B), B, 0, stream>>>(dinv, N);

  // --- layer 1: xw = x @ W1 (WMMA), aggregate, bias+ReLU ---
  const int waves = N / 16;  // 6250 row-tiles of 16
  gemm_wmma_f32<64><<<blocks((long long)waves * 32, B), B, 0, stream>>>(
      x, W1, xw, N);
  self_loop_init<64><<<blocks((long long)N * 64, B), B, 0, stream>>>(
      xw, dinv, h, N);
  edge_scatter<64><<<blocks((long long)E * 16, B), B, 0, stream>>>(
      src, dst, dinv, xw, h, E);
  bias_act<64, true><<<blocks((long long)N * 64, B), B, 0, stream>>>(h, b1, N);

  // --- layer 2: hw = h @ W2 (WMMA), aggregate into d_out, bias ---
  gemm_wmma_f32<32><<<blocks((long long)waves * 32, B), B, 0, stream>>>(
      h, W2, hw, N);
  self_loop_init<32><<<blocks((long long)N * 32, B), B, 0, stream>>>(
      hw, dinv, out, N);
  edge_scatter<32><<<blocks((long long)E * 8, B), B, 0, stream>>>(
      src, dst, dinv, hw, out, E);
  bias_act<32, false><<<blocks((long long)N * 32, B), B, 0, stream>>>(out, b2,
                                                                      N);
}